// FNO3DModel_3693671874577
// MI455X (gfx1250) — compile-verified
//
#include <hip/hip_runtime.h>
#include <math.h>

// ---------------------------------------------------------------------------
// FNO3D forward for MI455X (gfx1250, wave32, WMMA).
// Truncated separable DFTs (64 -> 8 modes/axis) replace full FFTs.
// f32 WMMA (v_wmma_f32_16x16x4_f32) drives all GEMM-shaped stages:
//   - fwd z-DFT (K=64)
//   - fused [inverse-z reconstruction (K=16) -> LDS transpose -> bypass conv
//     (K=32) + GELU], eliminating the 134 MB/layer sp round-trip
//   - proj1 (K=32) + GELU
// HBM-bound: ~11 GFLOP vs ~1.2 GB traffic @ 23.3 TB/s => ~50 us floor.
// ---------------------------------------------------------------------------

#define DD 64
#define NN (DD * DD * DD)   // 262144
#define BSZ 2
#define CC 32
#define PI_F 3.14159265358979f

typedef __attribute__((ext_vector_type(2))) float v2f;
typedef __attribute__((ext_vector_type(8))) float v8f;

// Workspace float offsets (total 26,345,600 floats ~= 100.5 MiB)
#define OFF_XA   ((size_t)0)          // x, [b][c][n]  (16,777,216)
#define OFF_Y1R  ((size_t)16777216)   // fwd-z re, [row=bc*4096+dx*64+dy][kz]
#define OFF_Y1I  ((size_t)18874368)
#define OFF_Y2R  ((size_t)20971520)   // fwd-y re, [(bc*64+dx)*8+ky][kz]
#define OFF_Y2I  ((size_t)21233664)
#define OFF_MDR  ((size_t)21495808)   // modes re, [b*32+c][m=kx*64+ky*8+kz]
#define OFF_MDI  ((size_t)21528576)
#define OFF_BLR  ((size_t)21561344)   // spectral-mixed modes
#define OFF_BLI  ((size_t)21594112)
#define OFF_U1R  ((size_t)21626880)   // inv-x out
#define OFF_U1I  ((size_t)21889024)
#define OFF_U2R  ((size_t)22151168)   // inv-y out, [row][kz]
#define OFF_U2I  ((size_t)24248320)
#define OFF_EMB  ((size_t)26345472)   // t_emb[64] then c_emb[64]

__device__ __forceinline__ v8f wmma_f32(v2f a, v2f b, v8f c) {
  // D(16x16,f32) = A(16x4,f32) * B(4x16,f32) + C
  return __builtin_amdgcn_wmma_f32_16x16x4_f32(false, a, false, b,
                                               (short)0, c, false, false);
}

// ---------------------------------------------------------------------------
// t_emb / c_emb: tiny (bs x 32) -- one thread per (b,c)
// ---------------------------------------------------------------------------
__global__ __launch_bounds__(64) void emb_kernel(
    const float* __restrict__ time_i, const float* __restrict__ cond,
    const float* __restrict__ tw, const float* __restrict__ tb,
    const float* __restrict__ cw, const float* __restrict__ cb,
    float* __restrict__ emb) {
  int t = threadIdx.x;
  if (t >= BSZ * CC) return;
  int b = t / CC, c = t % CC;

  float tv = time_i[b];
  float acc = tb[c];
  const float* w = tw + c * 11;
  float fr = PI_F;
  for (int i = 0; i < 5; i++) {
    float s, co; __sincosf(tv * fr, &s, &co);
    acc += co * w[i] + s * w[5 + i];
    fr *= 2.f;
  }
  acc += tv * w[10];
  emb[b * CC + c] = acc;

  float acc2 = cb[c];
  const float* w2 = cw + c * 352;
  for (int l = 0; l < 32; l++) {
    float v = cond[b * 32 + l];
    float fr2 = PI_F;
    for (int i = 0; i < 5; i++) {
      float s, co; __sincosf(v * fr2, &s, &co);
      acc2 += co * w2[l * 10 + i] + s * w2[l * 10 + 5 + i];
      fr2 *= 2.f;
    }
    acc2 += v * w2[320 + l];
  }
  emb[64 + b * CC + c] = acc2;
}

// ---------------------------------------------------------------------------
// lift: x[b][c][n] = lift_w @ node_feat + lift_b + t_emb + c_emb
// ---------------------------------------------------------------------------
__global__ __launch_bounds__(256) void lift_kernel(
    const float* __restrict__ state_in, const float* __restrict__ node_pos,
    const float* __restrict__ lw, const float* __restrict__ lb,
    const float* __restrict__ emb, float* __restrict__ xa) {
  __shared__ float w_s[CC * 37 + CC];
  for (int i = threadIdx.x; i < CC * 37; i += blockDim.x) w_s[i] = lw[i];
  for (int i = threadIdx.x; i < CC; i += blockDim.x) w_s[CC * 37 + i] = lb[i];
  __syncthreads();

  int idx = blockIdx.x * blockDim.x + threadIdx.x;   // bs*N = 524288
  int b = idx / NN, n = idx % NN;

  float f[37];
  const float4 st = *(const float4*)(state_in + (size_t)(b * NN + n) * 4);
  f[0] = st.x; f[1] = st.y; f[2] = st.z; f[3] = st.w;
  const float* pp = node_pos + (size_t)(b * NN + n) * 3;
#pragma unroll
  for (int l = 0; l < 3; l++) {
    float v = pp[l];
    float fr = PI_F;
#pragma unroll
    for (int i = 0; i < 5; i++) {
      float s, c; __sincosf(v * fr, &s, &c);
      f[4 + l * 10 + i] = c;
      f[4 + l * 10 + 5 + i] = s;
      fr *= 2.f;
    }
    f[34 + l] = v;
  }
  for (int c = 0; c < CC; c++) {
    float acc = w_s[CC * 37 + c] + emb[b * CC + c] + emb[64 + b * CC + c];
    const float* wr = w_s + c * 37;
#pragma unroll
    for (int k = 0; k < 37; k++) acc += f[k] * wr[k];
    xa[(size_t)(b * CC + c) * NN + n] = acc;
  }
}

// ---------------------------------------------------------------------------
// fwd z-DFT (WMMA): Y1[row][kz] = sum_z x[row][z] * e^{-2pi i kz z/64}
// M=16 rows/tile, K=64 (16 steps), N=16 = [re0..7 | im0..7].
// A-frag (ISA 7.12.2): lanes 0-15 K=k,k+1; lanes 16-31 K=k+2,k+3.
// B-frag assumed to mirror that K split (cols striped across lane halves).
// ---------------------------------------------------------------------------
__global__ __launch_bounds__(256) void fwdz_kernel(
    const float* __restrict__ xa, float* __restrict__ y1r,
    float* __restrict__ y1i) {
  int lane = threadIdx.x & 31;
  int wave = threadIdx.x >> 5;
  int tile = blockIdx.x * 8 + wave;          // 16384 tiles of 16 rows
  int row0 = tile * 16;
  int colh = lane & 15;
  int hi = lane >> 4;
  const float ang = -2.f * PI_F / 64.f;
  int kz = (colh < 8) ? colh : (colh - 8);

  // Wave-invariant B fragments; integer phase reduced mod 64 for accuracy.
  v2f bf[16];
#pragma unroll
  for (int kk = 0; kk < 16; kk++) {
    int k0 = kk * 4 + hi * 2, k1 = k0 + 1;
    float s0, c0, s1, c1;
    __sincosf(ang * (float)((kz * k0) & 63), &s0, &c0);
    __sincosf(ang * (float)((kz * k1) & 63), &s1, &c1);
    bf[kk].x = (colh < 8) ? c0 : s0;   // re cols: cos(th); im cols: sin(-th)
    bf[kk].y = (colh < 8) ? c1 : s1;
  }

  const float* arow = xa + (size_t)(row0 + colh) * 64;
  __builtin_prefetch(arow + 16 * 64, 0, 1);   // stream next tile
  v8f acc = {};
#pragma unroll
  for (int kk = 0; kk < 16; kk++) {
    v2f a = *(const v2f*)(arow + kk * 4 + hi * 2);   // global_load_b64
    acc = wmma_f32(a, bf[kk], acc);
  }
  int rbase = row0 + hi * 8;   // D: lanes 0-15 -> M=j, lanes 16-31 -> M=j+8
#pragma unroll
  for (int j = 0; j < 8; j++) {
    size_t r = (size_t)(rbase + j) * 8;
    if (colh < 8) y1r[r + colh] = acc[j];
    else          y1i[r + (colh - 8)] = acc[j];
  }
}

// ---------------------------------------------------------------------------
// fwd y-DFT (VALU+LDS twiddle table): Y2 = sum_dy Y1 * e^{-2pi i ky dy/64}
// ---------------------------------------------------------------------------
__global__ __launch_bounds__(256) void fwdy_kernel(
    const float* __restrict__ y1r, const float* __restrict__ y1i,
    float* __restrict__ y2r, float* __restrict__ y2i) {
  __shared__ float tc[64], ts[64];
  if (threadIdx.x < 64) {
    float s, c; __sincosf((2.f * PI_F / 64.f) * (float)threadIdx.x, &s, &c);
    tc[threadIdx.x] = c; ts[threadIdx.x] = s;
  }
  __syncthreads();
  int idx = blockIdx.x * blockDim.x + threadIdx.x;  // 262144
  int kz = idx & 7, ky = (idx >> 3) & 7, dx = (idx >> 6) & 63, bc = idx >> 12;
  size_t base = ((size_t)(bc * 64 + dx) * 64) * 8 + kz;
  float ar = 0.f, ai = 0.f;
  int p = 0;
  for (int dy = 0; dy < 64; dy++) {
    float c = tc[p], s = ts[p];
    float yr = y1r[base + (size_t)dy * 8];
    float yi = y1i[base + (size_t)dy * 8];
    ar += yr * c + yi * s;        // y * e^{-i th}
    ai += yi * c - yr * s;
    p = (p + ky) & 63;
  }
  size_t o = ((size_t)(bc * 64 + dx) * 8 + ky) * 8 + kz;
  y2r[o] = ar; y2i[o] = ai;
}

// ---------------------------------------------------------------------------
// fwd x-DFT (VALU+LDS): MD = sum_dx Y2 * e^{-2pi i kx dx/64}
// ---------------------------------------------------------------------------
__global__ __launch_bounds__(256) void fwdx_kernel(
    const float* __restrict__ y2r, const float* __restrict__ y2i,
    float* __restrict__ mdr, float* __restrict__ mdi) {
  __shared__ float tc[64], ts[64];
  if (threadIdx.x < 64) {
    float s, c; __sincosf((2.f * PI_F / 64.f) * (float)threadIdx.x, &s, &c);
    tc[threadIdx.x] = c; ts[threadIdx.x] = s;
  }
  __syncthreads();
  int idx = blockIdx.x * blockDim.x + threadIdx.x;  // 32768
  int kz = idx & 7, ky = (idx >> 3) & 7, kx = (idx >> 6) & 7, bc = idx >> 9;
  float ar = 0.f, ai = 0.f;
  int p = 0;
  for (int dx = 0; dx < 64; dx++) {
    float c = tc[p], s = ts[p];
    size_t q = ((size_t)(bc * 64 + dx) * 8 + ky) * 8 + kz;
    float yr = y2r[q], yi = y2i[q];
    ar += yr * c + yi * s;
    ai += yi * c - yr * s;
    p = (p + kx) & 63;
  }
  size_t o = (size_t)bc * 512 + ((kx * 8 + ky) * 8 + kz);
  mdr[o] = ar; mdi[o] = ai;
}

// ---------------------------------------------------------------------------
// spectral mix (VALU): BLK[b][o][m] = sum_i MD[b][i][m] * W[i][o][m] (complex)
// Per-mode weights -> not a GEMM; 8.4 MFLOP total.
// ---------------------------------------------------------------------------
__global__ __launch_bounds__(256) void spec_kernel(
    const float* __restrict__ mdr, const float* __restrict__ mdi,
    const float* __restrict__ wr, const float* __restrict__ wi,
    float* __restrict__ blr, float* __restrict__ bli) {
  int idx = blockIdx.x * blockDim.x + threadIdx.x;  // 32768
  int m = idx & 511, o = (idx >> 9) & 31, b = idx >> 14;
  float ar = 0.f, ai = 0.f;
  for (int i = 0; i < 32; i++) {
    float xr = mdr[(size_t)(b * 32 + i) * 512 + m];
    float xi = mdi[(size_t)(b * 32 + i) * 512 + m];
    float wrr = wr[(size_t)(i * 32 + o) * 512 + m];
    float wii = wi[(size_t)(i * 32 + o) * 512 + m];
    ar += xr * wrr - xi * wii;
    ai += xr * wii + xi * wrr;
  }
  size_t p = (size_t)(b * 32 + o) * 512 + m;
  blr[p] = ar; bli[p] = ai;
}

// ---------------------------------------------------------------------------
// inv x (VALU+LDS): U1 = (1/64) sum_kx BLK * e^{+2pi i kx dx/64}
// ---------------------------------------------------------------------------
__global__ __launch_bounds__(256) void invx_kernel(
    const float* __restrict__ blr, const float* __restrict__ bli,
    float* __restrict__ u1r, float* __restrict__ u1i) {
  __shared__ float tc[64], ts[64];
  if (threadIdx.x < 64) {
    float s, c; __sincosf((2.f * PI_F / 64.f) * (float)threadIdx.x, &s, &c);
    tc[threadIdx.x] = c; ts[threadIdx.x] = s;
  }
  __syncthreads();
  int idx = blockIdx.x * blockDim.x + threadIdx.x;  // 262144
  int kz = idx & 7, ky = (idx >> 3) & 7, dx = (idx >> 6) & 63, bc = idx >> 12;
  float ar = 0.f, ai = 0.f;
  int p = 0;
#pragma unroll
  for (int kx = 0; kx < 8; kx++) {
    float c = tc[p], s = ts[p];
    size_t q = (size_t)bc * 512 + ((kx * 8 + ky) * 8 + kz);
    float br = blr[q], bi = bli[q];
    ar += br * c - bi * s;        // y * e^{+i th}
    ai += br * s + bi * c;
    p = (p + dx) & 63;
  }
  size_t o = ((size_t)(bc * 64 + dx) * 8 + ky) * 8 + kz;
  u1r[o] = ar * (1.f / 64.f); u1i[o] = ai * (1.f / 64.f);
}

// ---------------------------------------------------------------------------
// inv y (VALU+LDS): U2[row][kz] = (1/64) sum_ky U1 * e^{+2pi i ky dy/64}
// ---------------------------------------------------------------------------
__global__ __launch_bounds__(256) void invy_kernel(
    const float* __restrict__ u1r, const float* __restrict__ u1i,
    float* __restrict__ u2r, float* __restrict__ u2i) {
  __shared__ float tc[64], ts[64];
  if (threadIdx.x < 64) {
    float s, c; __sincosf((2.f * PI_F / 64.f) * (float)threadIdx.x, &s, &c);
    tc[threadIdx.x] = c; ts[threadIdx.x] = s;
  }
  __syncthreads();
  int idx = blockIdx.x * blockDim.x + threadIdx.x;  // 2097152
  int kz = idx & 7, dy = (idx >> 3) & 63, dx = (idx >> 9) & 63, bc = idx >> 15;
  float ar = 0.f, ai = 0.f;
  int p = 0;
#pragma unroll
  for (int ky = 0; ky < 8; ky++) {
    float c = tc[p], s = ts[p];
    size_t q = ((size_t)(bc * 64 + dx) * 8 + ky) * 8 + kz;
    float br = u1r[q], bi = u1i[q];
    ar += br * c - bi * s;
    ai += br * s + bi * c;
    p = (p + dy) & 63;
  }
  size_t o = ((size_t)(bc * 4096 + dx * 64 + dy)) * 8 + kz;
  u2r[o] = ar * (1.f / 64.f); u2i[o] = ai * (1.f / 64.f);
}

// inverse-rfft z coefficient (Hermitian reconstruction, modes 0..7 only)
__device__ __forceinline__ float invz_coef(int k, int z) {
  const float w = 2.f * PI_F / 64.f;
  if (k == 0) return 1.f / 64.f;
  if (k < 8)  return (2.f / 64.f) * __cosf(w * (float)((k * z) & 63));
  if (k == 8) return 0.f;                       // Im of kz=0 discarded
  return (-2.f / 64.f) * __sinf(w * (float)(((k - 8) * z) & 63));
}

// ---------------------------------------------------------------------------
// FUSED inverse-z + bypass conv + GELU (WMMA + LDS transpose), in place on x.
// Tile = 16 consecutive z in one (b, dx, dy) row (== a 16-point n-tile).
// Step 1: invz GEMM (A rows = 32 channels of U2, K=16 -> 4 steps, two
//         16-row groups) produces sp tile with M=channel, N=z (8 WMMAs).
// Step 2: transpose via wave-private LDS patch (M=channel -> M=spatial).
// Step 3: C-frag preload = sp + bias; bypass GEMM over K=32 channels of x
//         (16 WMMAs); exact GELU; store back into x (reads precede writes
//         within the wave; tiles are disjoint across waves).
// ---------------------------------------------------------------------------
#define SPP 17   // LDS row pitch (16 + 1 to stagger banks)
__global__ __launch_bounds__(256) void invz_conv_gelu_kernel(
    const float* __restrict__ u2r, const float* __restrict__ u2i,
    float* __restrict__ x, const float* __restrict__ w,
    const float* __restrict__ bias) {
  __shared__ float sp_s[8][CC * SPP];
  int lane = threadIdx.x & 31;
  int wave = threadIdx.x >> 5;
  int tile = blockIdx.x * 8 + wave;        // 32768 tiles (bs*N/16)
  int b = tile >> 14;                      // 16384 tiles per batch
  int n0 = (tile & 16383) * 16;
  int zt = (n0 & 63) >> 4;                 // which 16-z slab
  int rowdxdy = n0 >> 6;                   // dx*64+dy
  int colh = lane & 15;
  int hi = lane >> 4;
  float* lds = sp_s[wave];

  // ---- Step 1+2: inverse-z for all 32 channels of this (dx,dy,zt) ----
#pragma unroll
  for (int g = 0; g < 2; g++) {            // channel groups 0-15, 16-31
    int c = g * 16 + colh;                 // A row = channel
    const float* ar = u2r + ((size_t)((b * 32 + c) * 4096 + rowdxdy)) * 8;
    const float* ai = u2i + ((size_t)((b * 32 + c) * 4096 + rowdxdy)) * 8;
    v8f acc = {};
#pragma unroll
    for (int kk = 0; kk < 4; kk++) {
      int k0 = kk * 4 + hi * 2;            // pairs never straddle re/im split
      v2f a = (k0 < 8) ? *(const v2f*)(ar + k0) : *(const v2f*)(ai + k0 - 8);
      int z = zt * 16 + colh;
      v2f bb;
      bb.x = invz_coef(k0, z);
      bb.y = invz_coef(k0 + 1, z);
      acc = wmma_f32(a, bb, acc);
    }
    // D element: M = channel-in-group (j + 8*hi), N = z_local (colh)
#pragma unroll
    for (int j = 0; j < 8; j++)
      lds[(g * 16 + hi * 8 + j) * SPP + colh] = acc[j];
  }
  // wave-private LDS patch + in-order DS ops within a wave: no barrier needed

  // ---- Step 3: bypass conv + GELU, C preloaded from transposed sp ----
  size_t xbase = (size_t)b * CC * NN;
  v8f acc0, acc1;
  {
    float bb0 = bias[colh], bb1 = bias[colh + 16];
#pragma unroll
    for (int j = 0; j < 8; j++) {          // C elem: col = o, M = spatial
      acc0[j] = lds[colh * SPP + hi * 8 + j] + bb0;
      acc1[j] = lds[(colh + 16) * SPP + hi * 8 + j] + bb1;
    }
  }
#pragma unroll
  for (int kk = 0; kk < 8; kk++) {
    int k0 = kk * 4 + hi * 2, k1 = k0 + 1;
    v2f a;
    a.x = x[xbase + (size_t)k0 * NN + n0 + colh];
    a.y = x[xbase + (size_t)k1 * NN + n0 + colh];
    v2f b0, b1;
    b0.x = w[colh * 32 + k0];              // w[o][c], o = colh (half 0)
    b0.y = w[colh * 32 + k1];
    b1.x = w[(colh + 16) * 32 + k0];       // half 1
    b1.y = w[(colh + 16) * 32 + k1];
    acc0 = wmma_f32(a, b0, acc0);
    acc1 = wmma_f32(a, b1, acc1);
  }
#pragma unroll
  for (int j = 0; j < 8; j++) {
    float v0 = acc0[j], v1 = acc1[j];
    v0 = 0.5f * v0 * (1.f + erff(v0 * 0.70710678118f));   // exact GELU
    v1 = 0.5f * v1 * (1.f + erff(v1 * 0.70710678118f));
    int n = n0 + hi * 8 + j;
    x[xbase + (size_t)colh * NN + n] = v0;
    x[xbase + (size_t)(colh + 16) * NN + n] = v1;
  }
}

// ---------------------------------------------------------------------------
// proj1: conv1x1 (32x32) + bias + GELU, in place on x (WMMA).
// ---------------------------------------------------------------------------
__global__ __launch_bounds__(256) void convgelu_kernel(
    float* __restrict__ x, const float* __restrict__ w,
    const float* __restrict__ bias) {
  int lane = threadIdx.x & 31;
  int wave = threadIdx.x >> 5;
  int tile = blockIdx.x * 8 + wave;        // 32768 tiles
  int b = tile >> 14;
  int n0 = (tile & 16383) * 16;
  int colh = lane & 15;
  int hi = lane >> 4;
  size_t xbase = (size_t)b * CC * NN;

  v8f acc0, acc1;
  {
    float bb0 = bias[colh], bb1 = bias[colh + 16];
#pragma unroll
    for (int j = 0; j < 8; j++) { acc0[j] = bb0; acc1[j] = bb1; }
  }
#pragma unroll
  for (int kk = 0; kk < 8; kk++) {
    int k0 = kk * 4 + hi * 2, k1 = k0 + 1;
    v2f a;
    a.x = x[xbase + (size_t)k0 * NN + n0 + colh];
    a.y = x[xbase + (size_t)k1 * NN + n0 + colh];
    v2f b0, b1;
    b0.x = w[colh * 32 + k0];
    b0.y = w[colh * 32 + k1];
    b1.x = w[(colh + 16) * 32 + k0];
    b1.y = w[(colh + 16) * 32 + k1];
    acc0 = wmma_f32(a, b0, acc0);
    acc1 = wmma_f32(a, b1, acc1);
  }
#pragma unroll
  for (int j = 0; j < 8; j++) {
    float v0 = acc0[j], v1 = acc1[j];
    v0 = 0.5f * v0 * (1.f + erff(v0 * 0.70710678118f));
    v1 = 0.5f * v1 * (1.f + erff(v1 * 0.70710678118f));
    int n = n0 + hi * 8 + j;
    x[xbase + (size_t)colh * NN + n] = v0;
    x[xbase + (size_t)(colh + 16) * NN + n] = v1;
  }
}

// ---------------------------------------------------------------------------
// proj2 (32 -> 4) + Euler step: out = state_in + DT * (W x + b)
// ---------------------------------------------------------------------------
__global__ __launch_bounds__(256) void proj2_kernel(
    const float* __restrict__ x, const float* __restrict__ w,
    const float* __restrict__ bias, const float* __restrict__ state_in,
    float* __restrict__ out) {
  int idx = blockIdx.x * blockDim.x + threadIdx.x;  // 524288
  int b = idx / NN, n = idx % NN;
  float a0 = bias[0], a1 = bias[1], a2 = bias[2], a3 = bias[3];
  size_t base = (size_t)b * CC * NN + n;
#pragma unroll 4
  for (int c = 0; c < CC; c++) {
    float xv = x[base + (size_t)c * NN];
    a0 += xv * w[c];
    a1 += xv * w[32 + c];
    a2 += xv * w[64 + c];
    a3 += xv * w[96 + c];
  }
  size_t ob = (size_t)(b * NN + n) * 4;
  const float4 st = *(const float4*)(state_in + ob);
  float4 ov;
  ov.x = st.x + 0.05f * a0;
  ov.y = st.y + 0.05f * a1;
  ov.z = st.z + 0.05f * a2;
  ov.w = st.w + 0.05f * a3;
  *(float4*)(out + ob) = ov;
}

// ---------------------------------------------------------------------------
extern "C" void kernel_launch(void* const* d_in, const int* in_sizes, int n_in,
                              void* d_out, int out_size, void* d_ws,
                              size_t ws_size, hipStream_t stream) {
  (void)in_sizes; (void)n_in; (void)out_size; (void)ws_size;
  const float* state_in   = (const float*)d_in[0];
  const float* node_pos   = (const float*)d_in[1];
  /* d_in[2] = edges (unused by reference) */
  const float* time_i     = (const float*)d_in[3];
  const float* conditions = (const float*)d_in[4];
  const float* t_embed_w  = (const float*)d_in[5];
  const float* t_embed_b  = (const float*)d_in[6];
  const float* c_embed_w  = (const float*)d_in[7];
  const float* c_embed_b  = (const float*)d_in[8];
  const float* lift_w     = (const float*)d_in[9];
  const float* lift_b     = (const float*)d_in[10];
  const float* spec_wr    = (const float*)d_in[11];
  const float* spec_wi    = (const float*)d_in[12];
  const float* byp_w      = (const float*)d_in[13];
  const float* byp_b      = (const float*)d_in[14];
  const float* proj1_w    = (const float*)d_in[15];
  const float* proj1_b    = (const float*)d_in[16];
  const float* proj2_w    = (const float*)d_in[17];
  const float* proj2_b    = (const float*)d_in[18];
  float* out = (float*)d_out;
  float* ws  = (float*)d_ws;

  float* XA  = ws + OFF_XA;
  float* Y1R = ws + OFF_Y1R; float* Y1I = ws + OFF_Y1I;
  float* Y2R = ws + OFF_Y2R; float* Y2I = ws + OFF_Y2I;
  float* MDR = ws + OFF_MDR; float* MDI = ws + OFF_MDI;
  float* BLR = ws + OFF_BLR; float* BLI = ws + OFF_BLI;
  float* U1R = ws + OFF_U1R; float* U1I = ws + OFF_U1I;
  float* U2R = ws + OFF_U2R; float* U2I = ws + OFF_U2I;
  float* EMB = ws + OFF_EMB;

  emb_kernel<<<1, 64, 0, stream>>>(time_i, conditions, t_embed_w, t_embed_b,
                                   c_embed_w, c_embed_b, EMB);
  lift_kernel<<<2048, 256, 0, stream>>>(state_in, node_pos, lift_w, lift_b,
                                        EMB, XA);
  for (int l = 0; l < 4; l++) {
    fwdz_kernel<<<2048, 256, 0, stream>>>(XA, Y1R, Y1I);
    fwdy_kernel<<<1024, 256, 0, stream>>>(Y1R, Y1I, Y2R, Y2I);
    fwdx_kernel<<<128, 256, 0, stream>>>(Y2R, Y2I, MDR, MDI);
    spec_kernel<<<128, 256, 0, stream>>>(MDR, MDI,
                                         spec_wr + (size_t)l * 32 * 32 * 512,
                                         spec_wi + (size_t)l * 32 * 32 * 512,
                                         BLR, BLI);
    invx_kernel<<<1024, 256, 0, stream>>>(BLR, BLI, U1R, U1I);
    invy_kernel<<<8192, 256, 0, stream>>>(U1R, U1I, U2R, U2I);
    invz_conv_gelu_kernel<<<4096, 256, 0, stream>>>(U2R, U2I, XA,
                                                    byp_w + l * 1024,
                                                    byp_b + l * 32);
  }
  convgelu_kernel<<<4096, 256, 0, stream>>>(XA, proj1_w, proj1_b);
  proj2_kernel<<<2048, 256, 0, stream>>>(XA, proj2_w, proj2_b, state_in, out);
}